// FusionMarkovGNN_53008486367387
// MI455X (gfx1250) — compile-verified
//
#include <hip/hip_runtime.h>
#include <hip/hip_bf16.h>

typedef __attribute__((ext_vector_type(2))) float v2f;
typedef __attribute__((ext_vector_type(8))) float v8f;

#define NB   4      // batch
#define CC   256    // channels
#define HWP  1024   // H*W
#define SS   16     // markov slices
#define KBIG 4096   // s*C

// passf[n][K][b] = 1.0 if all 16 rows of slice K have m[.,b] > 0 (pass column)
// reluf[n][K][b] = 1.0 if some-but-not-all rows positive (relu column)
// both 0.0 -> column zeroed.  sel(v) = pa*v + ra*max(v,0): pure arithmetic.
__global__ __launch_bounds__(256)
void markov_flags_kernel(const float* __restrict__ adj,
                         float* __restrict__ passf, float* __restrict__ reluf) {
  int idx = blockIdx.x * 256 + threadIdx.x;       // 0..16383
  int b   = idx & (CC - 1);
  int K   = (idx >> 8) & (SS - 1);
  int n   = idx >> 12;
  const float* m = adj + (size_t)n * CC * CC;
  int cnt = 0;
#pragma unroll
  for (int i = 0; i < 16; ++i)
    cnt += (m[(size_t)(K * 16 + i) * CC + b] > 0.0f) ? 1 : 0;
  passf[idx] = (cnt == 16) ? 1.0f : 0.0f;
  reluf[idx] = (cnt > 0 && cnt < 16) ? 1.0f : 0.0f;
}

// W_eff[l,n] = conv_w[l] @ mss[n]  ([256x4096]@[4096x256]), markov sel fused
// into the B-operand as FMA (no branches, EXEC stays all-ones).
// One wave per 32x32 output tile (2x2 register tiling, 4 accumulators).
__global__ __launch_bounds__(256)
void weff_kernel(const float* __restrict__ conv_w, const float* __restrict__ adj,
                 const float* __restrict__ passf, const float* __restrict__ reluf,
                 float* __restrict__ weff) {
  int lane = threadIdx.x & 31;
  int wave = (blockIdx.x * 256 + threadIdx.x) >> 5;   // 0..767
  int ln   = wave >> 6;                               // l*NB + n, 0..11
  int t    = wave & 63;                               // 8x8 tiles of 32x32
  int l = ln >> 2, n = ln & 3;
  int o0 = (t >> 3) << 5;
  int b0 = (t & 7) << 5;

  // CDNA5 f32 WMMA 16x16x4 striping:
  //   A: lane&15 = M; lane[4] selects K-pair {0,1} vs {2,3}
  //   B: lane&15 = N; same K-pair selection
  //   D: 8 VGPRs, M = 8*lane[4] + r, N = lane&15
  int row = lane & 15;
  int col = lane & 15;
  int kb  = (lane >> 4) << 1;                         // 0 or 2

  const float* cw0 = conv_w + (size_t)l * CC * KBIG + (size_t)(o0 + row) * KBIG;
  const float* cw1 = cw0 + (size_t)16 * KBIG;
  const float* mn0 = adj + (size_t)n * CC * CC + (b0 + col);
  const float* mn1 = mn0 + 16;
  const float* pf  = passf + n * KBIG + (b0 + col);
  const float* rf  = reluf + n * KBIG + (b0 + col);

  v8f acc00 = {}, acc01 = {}, acc10 = {}, acc11 = {};
  for (int K = 0; K < SS; ++K) {
    float pa0 = pf[K * CC],      ra0 = rf[K * CC];        // column b0+col
    float pa1 = pf[K * CC + 16], ra1 = rf[K * CC + 16];   // column b0+col+16
    const float* a0p = cw0 + K * CC;
    const float* a1p = cw1 + K * CC;
#pragma unroll 4
    for (int ch = 0; ch < 64; ++ch) {                 // 256 contraction / 4 per WMMA
      int a0 = ch * 4 + kb;
      v2f A0 = *(const v2f*)(a0p + a0);
      v2f A1 = *(const v2f*)(a1p + a0);
      float m00 = mn0[(size_t)a0 * CC], m01 = mn0[(size_t)(a0 + 1) * CC];
      float m10 = mn1[(size_t)a0 * CC], m11 = mn1[(size_t)(a0 + 1) * CC];
      v2f B0, B1;
      B0.x = pa0 * m00 + ra0 * fmaxf(m00, 0.0f);
      B0.y = pa0 * m01 + ra0 * fmaxf(m01, 0.0f);
      B1.x = pa1 * m10 + ra1 * fmaxf(m10, 0.0f);
      B1.y = pa1 * m11 + ra1 * fmaxf(m11, 0.0f);
      acc00 = __builtin_amdgcn_wmma_f32_16x16x4_f32(false, A0, false, B0, (short)0, acc00, false, false);
      acc01 = __builtin_amdgcn_wmma_f32_16x16x4_f32(false, A0, false, B1, (short)0, acc01, false, false);
      acc10 = __builtin_amdgcn_wmma_f32_16x16x4_f32(false, A1, false, B0, (short)0, acc10, false, false);
      acc11 = __builtin_amdgcn_wmma_f32_16x16x4_f32(false, A1, false, B1, (short)0, acc11, false, false);
    }
  }
  float* dst = weff + (size_t)ln * CC * CC;
  int mb = (lane >> 4) << 3;
#pragma unroll
  for (int r = 0; r < 8; ++r) {
    dst[(size_t)(o0 + mb + r)      * CC + (b0 + col)]      = acc00[r];
    dst[(size_t)(o0 + mb + r)      * CC + (b0 + col + 16)] = acc01[r];
    dst[(size_t)(o0 + 16 + mb + r) * CC + (b0 + col)]      = acc10[r];
    dst[(size_t)(o0 + 16 + mb + r) * CC + (b0 + col + 16)] = acc11[r];
  }
}

// y = relu(scale * (W_eff[l,n] @ x[n]) + bias), BN folded into scale/bias.
// One wave per 32x32 output tile of [256x1024] (2x2 register tiling).
__global__ __launch_bounds__(256)
void layer_kernel(const float* __restrict__ weff, const float* __restrict__ xin,
                  const float* __restrict__ conv_b, const float* __restrict__ gamma,
                  const float* __restrict__ beta, const float* __restrict__ mean,
                  const float* __restrict__ var, float* __restrict__ xout, int l) {
  int lane = threadIdx.x & 31;
  int wave = (blockIdx.x * 256 + threadIdx.x) >> 5;   // 0..1023
  int n    = wave >> 8;
  int t    = wave & 255;                              // 8 x 32 tiles of 32x32
  int o0 = (t >> 5) << 5;
  int p0 = (t & 31) << 5;

  int row = lane & 15;
  int col = lane & 15;
  int kb  = (lane >> 4) << 1;

  const float* W0 = weff + (size_t)(l * NB + n) * CC * CC + (size_t)(o0 + row) * CC;
  const float* W1 = W0 + (size_t)16 * CC;
  const float* X0 = xin + (size_t)n * CC * HWP + (p0 + col);
  const float* X1 = X0 + 16;

  v8f acc00 = {}, acc01 = {}, acc10 = {}, acc11 = {};
#pragma unroll 4
  for (int ch = 0; ch < 64; ++ch) {                   // K=256 contraction
    int k0 = ch * 4 + kb;
    v2f A0 = *(const v2f*)(W0 + k0);
    v2f A1 = *(const v2f*)(W1 + k0);
    v2f B0, B1;
    B0.x = X0[(size_t)k0 * HWP];
    B0.y = X0[(size_t)(k0 + 1) * HWP];
    B1.x = X1[(size_t)k0 * HWP];
    B1.y = X1[(size_t)(k0 + 1) * HWP];
    acc00 = __builtin_amdgcn_wmma_f32_16x16x4_f32(false, A0, false, B0, (short)0, acc00, false, false);
    acc01 = __builtin_amdgcn_wmma_f32_16x16x4_f32(false, A0, false, B1, (short)0, acc01, false, false);
    acc10 = __builtin_amdgcn_wmma_f32_16x16x4_f32(false, A1, false, B0, (short)0, acc10, false, false);
    acc11 = __builtin_amdgcn_wmma_f32_16x16x4_f32(false, A1, false, B1, (short)0, acc11, false, false);
  }

  int mb = (lane >> 4) << 3;
  float* outn = xout + (size_t)n * CC * HWP;
#pragma unroll
  for (int r = 0; r < 8; ++r) {
    int oa = o0 + mb + r;
    int ob = oa + 16;
    float sca = gamma[l * CC + oa] * rsqrtf(var[l * CC + oa] + 1e-5f);
    float bia = (conv_b[l * CC + oa] - mean[l * CC + oa]) * sca + beta[l * CC + oa];
    float scb = gamma[l * CC + ob] * rsqrtf(var[l * CC + ob] + 1e-5f);
    float bib = (conv_b[l * CC + ob] - mean[l * CC + ob]) * scb + beta[l * CC + ob];
    outn[(size_t)oa * HWP + (p0 + col)]      = fmaxf(acc00[r] * sca + bia, 0.0f);
    outn[(size_t)oa * HWP + (p0 + col + 16)] = fmaxf(acc01[r] * sca + bia, 0.0f);
    outn[(size_t)ob * HWP + (p0 + col)]      = fmaxf(acc10[r] * scb + bib, 0.0f);
    outn[(size_t)ob * HWP + (p0 + col + 16)] = fmaxf(acc11[r] * scb + bib, 0.0f);
  }
}

extern "C" void kernel_launch(void* const* d_in, const int* in_sizes, int n_in,
                              void* d_out, int out_size, void* d_ws, size_t ws_size,
                              hipStream_t stream) {
  const float* feats  = (const float*)d_in[0];
  const float* adj    = (const float*)d_in[1];
  const float* conv_w = (const float*)d_in[2];
  const float* conv_b = (const float*)d_in[3];
  const float* gamma  = (const float*)d_in[4];
  const float* beta   = (const float*)d_in[5];
  const float* mean   = (const float*)d_in[6];
  const float* var    = (const float*)d_in[7];
  float* out = (float*)d_out;

  float* passf = (float*)d_ws;                               // 64 KB
  float* reluf = (float*)((char*)d_ws + (64 << 10));         // 64 KB
  float* weff  = (float*)((char*)d_ws + (128 << 10));        // 3 MB: [L*N][256][256]

  // 1) markov column multipliers (arithmetic select, no branches downstream)
  markov_flags_kernel<<<dim3(64), dim3(256), 0, stream>>>(adj, passf, reluf);

  // 2) W_eff = conv_w @ mss (layer-invariant): 768 waves, 32x32 tile each
  weff_kernel<<<dim3(96), dim3(256), 0, stream>>>(conv_w, adj, passf, reluf, weff);

  // 3) three chained layer GEMMs with fused BN+ReLU epilogue
  const float* xin = feats;
  for (int l = 0; l < 3; ++l) {
    float* xout = out + (size_t)l * NB * CC * HWP;
    layer_kernel<<<dim3(128), dim3(256), 0, stream>>>(weff, xin, conv_b, gamma,
                                                      beta, mean, var, xout, l);
    xin = xout;
  }
}